// VP_Attention_Module_xlan_simple_fusion_h_50586124812545
// MI455X (gfx1250) — compile-verified
//
#include <hip/hip_runtime.h>

typedef __attribute__((ext_vector_type(2))) float v2f;
typedef __attribute__((ext_vector_type(4))) float f4;
typedef __attribute__((ext_vector_type(8))) float v8f;

#define Bsz  64
#define Mlen 1024
#define E    1024
#define NH   8
#define HD   128
#define MID  64
#define ALPHA_C   1.3f
#define RALPHA_C  (1.0f / 1.3f)
#define WBP  72    // padded LDS stride for Wb (half-wave bank ranges disjoint)
#define KSP  132   // padded LDS stride for key tiles (conflict-free A reads)

__device__ __forceinline__ float celu_f(float x) {
    return fmaxf(x, 0.f) + fminf(0.f, ALPHA_C * (__expf(x * RALPHA_C) - 1.f));
}

__device__ __forceinline__ v8f wmma_f32(v2f a, v2f b, v8f c) {
    // D = A(16x4) * B(4x16) + C, fp32, wave32
    return __builtin_amdgcn_wmma_f32_16x16x4_f32(false, a, false, b, (short)0, c, false, false);
}

// async global -> LDS copy, 16B per lane (ASYNCcnt-tracked)
__device__ __forceinline__ void async_copy_b128(unsigned lds_off, const void* gaddr) {
    asm volatile("global_load_async_to_lds_b128 %0, %1, off"
                 :: "v"(lds_off), "v"((unsigned long long)gaddr) : "memory");
}
__device__ __forceinline__ void wait_async0() {
    asm volatile("s_wait_asynccnt 0x0" ::: "memory");
}

// ---------------------------------------------------------------------------
// K1: X = celu(h_state @ W + bias), 16x16 tile per wave, K=1024 in steps of 4
// grid: (E/16, B/16, 2)   block: 32
// ---------------------------------------------------------------------------
__global__ __launch_bounds__(32)
void k_in_gemm(const float* __restrict__ h,
               const float* __restrict__ Wq, const float* __restrict__ bq,
               const float* __restrict__ Wv, const float* __restrict__ bv,
               float* __restrict__ xq, float* __restrict__ xv) {
    const int lane = threadIdx.x;
    const int l15 = lane & 15, half = lane >> 4;
    const int n0 = blockIdx.x * 16, r0 = blockIdx.y * 16;
    const float* W    = blockIdx.z ? Wv : Wq;
    const float* bias = blockIdx.z ? bv : bq;
    float*       X    = blockIdx.z ? xv : xq;

    const int rA = r0 + l15;          // A row for this lane
    const int kof = half * 2;         // A/B k sub-offset for this lane half
    const int nB = n0 + l15;          // B/C column for this lane

    v8f acc = {};
#pragma unroll 4
    for (int k = 0; k < E; k += 4) {
        v2f a, b;
        a.x = h[(size_t)rA * E + k + kof];
        a.y = h[(size_t)rA * E + k + kof + 1];
        b.x = W[(size_t)(k + kof) * E + nB];
        b.y = W[(size_t)(k + kof + 1) * E + nB];
        acc = wmma_f32(a, b, acc);
    }
    const float bn = bias[nB];
    const int rowbase = r0 + half * 8;
#pragma unroll
    for (int g = 0; g < 8; ++g)
        X[(size_t)(rowbase + g) * E + nB] = celu_f(acc[g] + bn);
}

// ---------------------------------------------------------------------------
// K2: group norm (per row, per head group of 128 channels, ddof=1)
// grid: (B*NH, 2)  block: 128
// ---------------------------------------------------------------------------
__global__ __launch_bounds__(128)
void k_gn(const float* __restrict__ xq, const float* __restrict__ xv,
          const float* __restrict__ gwq, const float* __restrict__ gbq,
          const float* __restrict__ gwv, const float* __restrict__ gbv,
          float* __restrict__ q, float* __restrict__ v1) {
    __shared__ float s[128];
    const int t = threadIdx.x;
    const int b = blockIdx.x >> 3, g = blockIdx.x & 7;
    const float* X  = blockIdx.y ? xv  : xq;
    const float* gw = blockIdx.y ? gwv : gwq;
    const float* gb = blockIdx.y ? gbv : gbq;
    float*       O  = blockIdx.y ? v1  : q;

    const int c = g * HD + t;
    const float x = X[(size_t)b * E + c];
    s[t] = x; __syncthreads();
    for (int o = 64; o; o >>= 1) { if (t < o) s[t] += s[t + o]; __syncthreads(); }
    const float mean = s[0] * (1.f / 128.f);
    __syncthreads();
    const float d = x - mean;
    s[t] = d * d; __syncthreads();
    for (int o = 64; o; o >>= 1) { if (t < o) s[t] += s[t + o]; __syncthreads(); }
    const float var = s[0] * (1.f / 127.f);
    O[(size_t)b * E + c] = d * rsqrtf(var + 1e-5f) * gw[c] + gb[c];
}

// ---------------------------------------------------------------------------
// K3: fused per-(b,h) attention.  Key tiles streamed with async global->LDS
// copies (no VGPR round-trip); q folded into Wb rows (key@(diag(q)Wb)).
// grid: (B*NH)  block: 256 (8 waves; each wave owns 16-row chunks)
// ---------------------------------------------------------------------------
__global__ __launch_bounds__(256)
void k_attn(const float* __restrict__ paf, const float* __restrict__ mask,
            const float* __restrict__ q, const float* __restrict__ v1,
            const float* __restrict__ Wb, const float* __restrict__ bb,
            const float* __restrict__ Ws, const float* __restrict__ bs,
            const float* __restrict__ Wc, const float* __restrict__ bc,
            float* __restrict__ att) {
    __shared__ __align__(16) float Wb_sh[HD * WBP];      // 36.9 KB (q-scaled rows)
    __shared__ __align__(16) float key_sh[8][16 * KSP];  // 67.5 KB
    __shared__ float logits_sh[8][16];
    __shared__ __align__(16) float wsum_sh[8][HD];
    __shared__ float pool_sh[8][MID];
    __shared__ float maxd_sh[8], den_sh[8], msum_sh[8];
    __shared__ float poolf[MID];

    const int tid = threadIdx.x;
    const int wave = tid >> 5, lane = tid & 31;
    const int l15 = lane & 15, half = lane >> 4;
    const int b = blockIdx.x >> 3, h = blockIdx.x & 7;

    // stage Wb[h] (128x64) into padded LDS, rows pre-scaled by q[b,h,:]
    for (int i = tid; i < HD * MID; i += 256) {
        const int k = i >> 6, n = i & 63;
        Wb_sh[k * WBP + n] = Wb[(size_t)h * HD * MID + i] * q[(size_t)b * E + h * HD + k];
    }
    __syncthreads();

    float wsv[4], bbv[4];
#pragma unroll
    for (int t = 0; t < 4; ++t) {
        wsv[t] = Ws[h * MID + t * 16 + l15];
        bbv[t] = bb[h * MID + t * 16 + l15];
    }
    const float bsv = bs[h];

    float runmax = -__builtin_inff(), denom = 0.f, msum = 0.f;
    f4 wsum = {0.f, 0.f, 0.f, 0.f};
    float pool[4] = {0.f, 0.f, 0.f, 0.f};
    float* ksh = &key_sh[wave][0];
    const unsigned ksh_off = (unsigned)(uintptr_t)(void*)ksh + (unsigned)(lane * 16);

    for (int it = 0; it < 8; ++it) {
        const int m0 = (it * 8 + wave) * 16;
        // async-stage raw key tile: one b128 async op moves a full 512B row
        const char* src = (const char*)(paf + ((size_t)(b * Mlen + m0)) * (2 * E) + h * HD)
                          + (size_t)(lane * 16);
#pragma unroll
        for (int i = 0; i < 16; ++i)
            async_copy_b128(ksh_off + (unsigned)(i * KSP * 4), src + (size_t)i * (2 * E * 4));
        wait_async0();

        v8f acc[4] = {{}, {}, {}, {}};
#pragma unroll 4
        for (int ks = 0; ks < 32; ++ks) {
            const int kk = ks * 4 + half * 2;
            v2f a;
            a.x = ksh[l15 * KSP + kk];
            a.y = ksh[l15 * KSP + kk + 1];
#pragma unroll
            for (int t = 0; t < 4; ++t) {
                v2f bv;
                bv.x = Wb_sh[kk * WBP + t * 16 + l15];
                bv.y = Wb_sh[(kk + 1) * WBP + t * 16 + l15];
                acc[t] = wmma_f32(a, bv, acc[t]);
            }
        }
        // bias + relu in registers (am), masked pool, logits = am @ Ws
#pragma unroll
        for (int t = 0; t < 4; ++t)
#pragma unroll
            for (int g = 0; g < 8; ++g)
                acc[t][g] = fmaxf(acc[t][g] + bbv[t], 0.f);
#pragma unroll
        for (int g = 0; g < 8; ++g) {
            const float mg = mask[(size_t)b * Mlen + m0 + g + half * 8];
#pragma unroll
            for (int t = 0; t < 4; ++t) pool[t] += acc[t][g] * mg;
            float p = acc[0][g] * wsv[0] + acc[1][g] * wsv[1] +
                      acc[2][g] * wsv[2] + acc[3][g] * wsv[3];
            p += __shfl_xor(p, 1); p += __shfl_xor(p, 2);
            p += __shfl_xor(p, 4); p += __shfl_xor(p, 8);
            if (l15 == 0) logits_sh[wave][half * 8 + g] = p;
        }
        // online softmax + weighted p_value2 accumulation (one pass)
        for (int j = 0; j < 16; ++j) {
            const int m = m0 + j;
            const float mg = mask[(size_t)b * Mlen + m];
            float l = logits_sh[wave][j] + bsv;
            if (mg == 0.f) l = -1e9f;
            msum += mg;
            const float nm = fmaxf(runmax, l);
            const float sc = __expf(runmax - nm);
            const float ex = __expf(l - nm);
            runmax = nm;
            denom = denom * sc + ex;
            const f4 v2 = *(const f4*)(paf + ((size_t)(b * Mlen + m)) * (2 * E)
                                       + E + h * HD + lane * 4);
            wsum = wsum * sc + v2 * ex;
        }
    }
    // per-wave partials -> LDS
    if (lane == 0) { maxd_sh[wave] = runmax; den_sh[wave] = denom; msum_sh[wave] = msum; }
    *(f4*)&wsum_sh[wave][lane * 4] = wsum;
#pragma unroll
    for (int t = 0; t < 4; ++t) pool[t] += __shfl_xor(pool[t], 16);
    if (lane < 16)
#pragma unroll
        for (int t = 0; t < 4; ++t) pool_sh[wave][t * 16 + lane] = pool[t];
    __syncthreads();

    if (tid < MID) {
        float ps = 0.f, ms = 0.f;
#pragma unroll
        for (int w = 0; w < 8; ++w) { ps += pool_sh[w][tid]; ms += msum_sh[w]; }
        poolf[tid] = ps / ms;
    }
    __syncthreads();

    if (tid < HD) {
        float G = -__builtin_inff();
#pragma unroll
        for (int w = 0; w < 8; ++w) G = fmaxf(G, maxd_sh[w]);
        float D = 0.f, wv = 0.f;
#pragma unroll
        for (int w = 0; w < 8; ++w) {
            const float e = __expf(maxd_sh[w] - G);
            D += e * den_sh[w];
            wv += e * wsum_sh[w][tid];
        }
        const float attv = wv / D;
        float a = bc[h * HD + tid];
        for (int j = 0; j < MID; ++j)
            a += poolf[j] * Wc[(size_t)(h * MID + j) * HD + tid];
        const float gate = 1.f / (1.f + __expf(-a));
        att[(size_t)b * E + h * HD + tid] =
            attv * v1[(size_t)b * E + h * HD + tid] * gate;
    }
}

// ---------------------------------------------------------------------------
// K4: out_raw = concat([h_state, att]) @ Wt + bt  (K=2048, WMMA tiles)
// grid: (E/16, B/16)  block: 32
// ---------------------------------------------------------------------------
__global__ __launch_bounds__(32)
void k_out_gemm(const float* __restrict__ h, const float* __restrict__ att,
                const float* __restrict__ Wt, const float* __restrict__ bt,
                float* __restrict__ outr) {
    const int lane = threadIdx.x;
    const int l15 = lane & 15, half = lane >> 4;
    const int n0 = blockIdx.x * 16, r0 = blockIdx.y * 16;
    const int rA = r0 + l15, kof = half * 2, nB = n0 + l15;

    v8f acc = {};
#pragma unroll 4
    for (int k = 0; k < 2 * E; k += 4) {
        const int kk = k + kof;
        const float* src = (kk < E) ? (h + (size_t)rA * E + kk)
                                    : (att + (size_t)rA * E + (kk - E));
        v2f a, b;
        a.x = src[0]; a.y = src[1];
        b.x = Wt[(size_t)kk * E + nB];
        b.y = Wt[(size_t)(kk + 1) * E + nB];
        acc = wmma_f32(a, b, acc);
    }
    const float bn = bt[nB];
    const int rowbase = r0 + half * 8;
#pragma unroll
    for (int g = 0; g < 8; ++g)
        outr[(size_t)(rowbase + g) * E + nB] = acc[g] + bn;
}

// ---------------------------------------------------------------------------
// K5: LayerNorm over last dim (ddof=0)
// grid: B   block: 256 (4 elems/thread)
// ---------------------------------------------------------------------------
__global__ __launch_bounds__(256)
void k_ln(const float* __restrict__ outr, const float* __restrict__ lnw,
          const float* __restrict__ lnb, float* __restrict__ out) {
    __shared__ float s[256];
    const int t = threadIdx.x, r = blockIdx.x;
    const f4 x = *(const f4*)(outr + (size_t)r * E + t * 4);
    s[t] = x.x + x.y + x.z + x.w; __syncthreads();
    for (int o = 128; o; o >>= 1) { if (t < o) s[t] += s[t + o]; __syncthreads(); }
    const float mean = s[0] * (1.f / (float)E);
    __syncthreads();
    f4 d; d.x = x.x - mean; d.y = x.y - mean; d.z = x.z - mean; d.w = x.w - mean;
    s[t] = d.x * d.x + d.y * d.y + d.z * d.z + d.w * d.w; __syncthreads();
    for (int o = 128; o; o >>= 1) { if (t < o) s[t] += s[t + o]; __syncthreads(); }
    const float rstd = rsqrtf(s[0] * (1.f / (float)E) + 1e-5f);
#pragma unroll
    for (int j = 0; j < 4; ++j) {
        const int c = t * 4 + j;
        out[(size_t)r * E + c] = d[j] * rstd * lnw[c] + lnb[c];
    }
}

// ---------------------------------------------------------------------------
extern "C" void kernel_launch(void* const* d_in, const int* in_sizes, int n_in,
                              void* d_out, int out_size, void* d_ws, size_t ws_size,
                              hipStream_t stream) {
    const float* h    = (const float*)d_in[0];
    // d_in[1] (att_feats) is unused by the reference
    const float* mask = (const float*)d_in[2];
    const float* paf  = (const float*)d_in[3];
    const float* Wq   = (const float*)d_in[4];
    const float* bq   = (const float*)d_in[5];
    const float* gwq  = (const float*)d_in[6];
    const float* gbq  = (const float*)d_in[7];
    const float* Wv1  = (const float*)d_in[8];
    const float* bv1  = (const float*)d_in[9];
    const float* gwv  = (const float*)d_in[10];
    const float* gbv  = (const float*)d_in[11];
    const float* Wb   = (const float*)d_in[12];
    const float* bb   = (const float*)d_in[13];
    const float* Ws   = (const float*)d_in[14];
    const float* bs   = (const float*)d_in[15];
    const float* Wc   = (const float*)d_in[16];
    const float* bc   = (const float*)d_in[17];
    const float* Wt   = (const float*)d_in[18];
    const float* bt   = (const float*)d_in[19];
    const float* lnw  = (const float*)d_in[20];
    const float* lnb  = (const float*)d_in[21];

    float* ws   = (float*)d_ws;
    const size_t S = (size_t)Bsz * E;   // 65536
    float* xq   = ws + 0 * S;
    float* xv   = ws + 1 * S;
    float* q    = ws + 2 * S;
    float* v1   = ws + 3 * S;
    float* att  = ws + 4 * S;
    float* outr = ws + 5 * S;

    k_in_gemm<<<dim3(E / 16, Bsz / 16, 2), 32, 0, stream>>>(h, Wq, bq, Wv1, bv1, xq, xv);
    k_gn<<<dim3(Bsz * NH, 2), 128, 0, stream>>>(xq, xv, gwq, gbq, gwv, gbv, q, v1);
    k_attn<<<dim3(Bsz * NH), 256, 0, stream>>>(paf, mask, q, v1, Wb, bb, Ws, bs, Wc, bc, att);
    k_out_gemm<<<dim3(E / 16, Bsz / 16), 32, 0, stream>>>(h, att, Wt, bt, outr);
    k_ln<<<dim3(Bsz), 256, 0, stream>>>(outr, lnw, lnb, (float*)d_out);
}